// LSTM_2370821948014
// MI455X (gfx1250) — compile-verified
//
#include <hip/hip_runtime.h>
#include <math.h>
#include <stdint.h>

typedef float v2f __attribute__((ext_vector_type(2)));
typedef float v8f __attribute__((ext_vector_type(8)));
typedef int   v4i __attribute__((ext_vector_type(4)));
typedef __attribute__((address_space(3))) v4i v4i_lds;

#define Bsz 65536
#define Isz 128
#define Hsz 256
#define Osz 256
#define KD  384      // I + H
#define BM  64       // batch rows per workgroup
#define LDA 388      // padded LDS stride for combined tile (floats)
#define LDP 260      // padded LDS stride for logits tile (floats)

// ---- optional CDNA5 async global->LDS path (guarded: falls back cleanly) ----
#if defined(__has_builtin)
#  if __has_builtin(__builtin_amdgcn_global_load_async_to_lds_b128)
#    define USE_ASYNC_LDS 1
#  endif
#  if __has_builtin(__builtin_amdgcn_s_wait_asynccnt)
#    define HAVE_WAIT_ASYNCCNT 1
#  endif
#endif

__device__ __forceinline__ float fast_sigmoid(float x) {
    return 1.0f / (1.0f + __expf(-x));
}
__device__ __forceinline__ float fast_tanh(float x) {
    float e = __expf(-2.0f * fabsf(x));
    float r = (1.0f - e) / (1.0f + e);
    return copysignf(r, x);
}
__device__ __forceinline__ v8f wmma_f32(v2f a, v2f b, v8f c) {
    // D = A(16x4,f32) * B(4x16,f32) + C(16x16,f32)
    return __builtin_amdgcn_wmma_f32_16x16x4_f32(false, a, false, b, (short)0, c,
                                                 false, false);
}

__global__ __launch_bounds__(256)
void lstm_fused_kernel(const float* __restrict__ input,
                       const float* __restrict__ hidden,
                       const float* __restrict__ w_i2h,
                       const float* __restrict__ b_i2h,
                       const float* __restrict__ w_i2o,
                       const float* __restrict__ b_i2o,
                       float* __restrict__ out)   // [B*O logsoftmax | B*H new_hidden]
{
    extern __shared__ float smem[];
    float* A    = smem;                 // BM x LDA  (combined tile)
    float* P    = A    + BM * LDA;      // BM x LDP  (logits tile)
    float* redA = P    + BM * LDP;      // BM x 4
    float* redB = redA + BM * 4;        // BM x 4
    float* lse  = redB + BM * 4;        // BM

    const int tid  = threadIdx.x;
    const int wv   = tid >> 5;          // wave id 0..7
    const int lane = tid & 31;
    const int sub  = lane >> 4;         // lane-group (0: lanes 0-15, 1: 16-31)
    const int l16  = lane & 15;
    const int row0 = blockIdx.x * BM;

    // ---- Stage combined = [input | hidden] tile into LDS ----
#ifdef USE_ASYNC_LDS
    {
        const float* in_base = input + (size_t)row0 * Isz;
        #pragma unroll
        for (int i = 0; i < (BM * Isz / 4) / 256; ++i) {    // 8 async b128 per thread
            int f = i * 256 + tid;
            int r = f >> 5;                                  // 32 float4 per row
            int c = f & 31;
            __builtin_amdgcn_global_load_async_to_lds_b128(
                (v4i*)(uintptr_t)(in_base + r * Isz + c * 4),
                (v4i_lds*)(uintptr_t)(&A[r * LDA + c * 4]), 0, 0);
        }
        const float* hd_base = hidden + (size_t)row0 * Hsz;
        #pragma unroll
        for (int i = 0; i < (BM * Hsz / 4) / 256; ++i) {    // 16 async b128 per thread
            int f = i * 256 + tid;
            int r = f >> 6;                                  // 64 float4 per row
            int c = f & 63;
            __builtin_amdgcn_global_load_async_to_lds_b128(
                (v4i*)(uintptr_t)(hd_base + r * Hsz + c * 4),
                (v4i_lds*)(uintptr_t)(&A[r * LDA + Isz + c * 4]), 0, 0);
        }
    }
#  ifdef HAVE_WAIT_ASYNCCNT
    __builtin_amdgcn_s_wait_asynccnt(0);
#  else
    asm volatile("s_wait_asynccnt 0x0" ::: "memory");
#  endif
#else
    {
        const float4* in4 = reinterpret_cast<const float4*>(input + (size_t)row0 * Isz);
        #pragma unroll
        for (int i = 0; i < (BM * Isz / 4) / 256; ++i) {    // 8 iters
            int f = i * 256 + tid;
            int r = f >> 5;                                  // 32 float4 per row
            int c = f & 31;
            *reinterpret_cast<float4*>(&A[r * LDA + c * 4]) = in4[r * 32 + c];
        }
        const float4* hd4 = reinterpret_cast<const float4*>(hidden + (size_t)row0 * Hsz);
        #pragma unroll
        for (int i = 0; i < (BM * Hsz / 4) / 256; ++i) {    // 16 iters
            int f = i * 256 + tid;
            int r = f >> 6;                                  // 64 float4 per row
            int c = f & 63;
            *reinterpret_cast<float4*>(&A[r * LDA + Isz + c * 4]) = hd4[r * 64 + c];
        }
    }
#endif
    __syncthreads();

    // ---- GEMM1: gates = combined @ w_i2h^T, fused LSTM cell update ----
    // Each wave-task: one 16-row M tile x one 16-col j tile, all 4 gates at once.
    float* out_nh = out + (size_t)Bsz * Osz;
    for (int it = 0; it < 8; ++it) {
        int task = wv + (it << 3);
        int m    = task & 3;            // M tile 0..3
        int jt   = task >> 2;           // j tile 0..15
        const int j = jt * 16 + l16;    // this lane's hidden column
        v8f ch = {}, ci = {}, cf = {}, co = {};
        const float* a_base = &A[(m * 16 + l16) * LDA + 2 * sub];
        const float* wb0 = w_i2h + ((size_t)0 * Hsz + j) * KD + 2 * sub;
        const float* wb1 = wb0 + (size_t)Hsz * KD;
        const float* wb2 = wb1 + (size_t)Hsz * KD;
        const float* wb3 = wb2 + (size_t)Hsz * KD;
        #pragma unroll 4
        for (int k = 0; k < KD; k += 4) {
            v2f a  = *reinterpret_cast<const v2f*>(a_base + k);
            v2f b0 = *reinterpret_cast<const v2f*>(wb0 + k);
            v2f b1 = *reinterpret_cast<const v2f*>(wb1 + k);
            v2f b2 = *reinterpret_cast<const v2f*>(wb2 + k);
            v2f b3 = *reinterpret_cast<const v2f*>(wb3 + k);
            ch = wmma_f32(a, b0, ch);
            ci = wmma_f32(a, b1, ci);
            cf = wmma_f32(a, b2, cf);
            co = wmma_f32(a, b3, co);
        }
        float bh = b_i2h[j];
        float bi = b_i2h[Hsz + j];
        float bf = b_i2h[2 * Hsz + j];
        float bo = b_i2h[3 * Hsz + j];
        #pragma unroll
        for (int r = 0; r < 8; ++r) {
            int lrow  = m * 16 + r + 8 * sub;       // C layout: M = r + 8*sub
            float hg  = fast_tanh(ch[r] + bh);
            float igs = fast_sigmoid(ci[r] + bi);
            float fgs = fast_sigmoid(cf[r] + bf);
            float ogs = fast_sigmoid(co[r] + bo);
            float hv  = A[lrow * LDA + Isz + j];    // old hidden from LDS tile
            float nh  = ogs * fast_tanh(fgs * hv + igs * hg);
            out_nh[(size_t)(row0 + lrow) * Hsz + j] = nh;
        }
    }

    // ---- GEMM2: logits = combined @ w_i2o^T + b, staged to LDS ----
    for (int it = 0; it < 8; ++it) {
        int task = wv + (it << 3);
        int m    = task & 3;
        int nt   = task >> 2;
        const int n = nt * 16 + l16;
        v8f c = {};
        const float* a_base = &A[(m * 16 + l16) * LDA + 2 * sub];
        const float* wb = w_i2o + (size_t)n * KD + 2 * sub;
        #pragma unroll 4
        for (int k = 0; k < KD; k += 4) {
            v2f a = *reinterpret_cast<const v2f*>(a_base + k);
            v2f b = *reinterpret_cast<const v2f*>(wb + k);
            c = wmma_f32(a, b, c);
        }
        float bo = b_i2o[n];
        #pragma unroll
        for (int r = 0; r < 8; ++r) {
            int lrow = m * 16 + r + 8 * sub;
            P[lrow * LDP + n] = c[r] + bo;
        }
    }
    __syncthreads();

    // ---- log-softmax across 256 cols per row (4 threads / row) ----
    {
        int row = tid >> 2;
        int q   = tid & 3;
        const float* prow = &P[row * LDP + q * 64];
        float mx = -INFINITY;
        #pragma unroll 4
        for (int c = 0; c < 64; ++c) mx = fmaxf(mx, prow[c]);
        redA[row * 4 + q] = mx;
        __syncthreads();
        mx = fmaxf(fmaxf(redA[row * 4 + 0], redA[row * 4 + 1]),
                   fmaxf(redA[row * 4 + 2], redA[row * 4 + 3]));
        float s = 0.0f;
        #pragma unroll 4
        for (int c = 0; c < 64; ++c) s += __expf(prow[c] - mx);
        redB[row * 4 + q] = s;
        __syncthreads();
        s = (redB[row * 4 + 0] + redB[row * 4 + 1]) +
            (redB[row * 4 + 2] + redB[row * 4 + 3]);
        if (q == 0) lse[row] = mx + __logf(s);
    }
    __syncthreads();

    // ---- coalesced output store: thread = column ----
    for (int i = 0; i < BM; ++i) {
        out[(size_t)(row0 + i) * Osz + tid] = P[i * LDP + tid] - lse[i];
    }
}

extern "C" void kernel_launch(void* const* d_in, const int* in_sizes, int n_in,
                              void* d_out, int out_size, void* d_ws, size_t ws_size,
                              hipStream_t stream) {
    const float* input  = (const float*)d_in[0];
    const float* hidden = (const float*)d_in[1];
    const float* w_i2h  = (const float*)d_in[2];
    const float* b_i2h  = (const float*)d_in[3];
    const float* w_i2o  = (const float*)d_in[4];
    const float* b_i2o  = (const float*)d_in[5];
    float* out = (float*)d_out;

    size_t shmem = (size_t)(BM * LDA + BM * LDP + BM * 8 + BM) * sizeof(float);
    dim3 grid(Bsz / BM);
    lstm_fused_kernel<<<grid, 256, shmem, stream>>>(input, hidden, w_i2h, b_i2h,
                                                    w_i2o, b_i2o, out);
}